// InteractionBlock_50697793962049
// MI455X (gfx1250) — compile-verified
//
#include <hip/hip_runtime.h>
#include <hip/hip_bf16.h>
#include <math.h>

// ---------------------------------------------------------------------------
// MACE-like interaction block for MI455X (gfx1250), wave32, WMMA f32 16x16x4.
// Memory-bound workload: all GEMMs done in full f32 via V_WMMA_F32_16X16X4_F32
// (precision-preserving; matrix throughput is not the bottleneck).
// ---------------------------------------------------------------------------

typedef float v2f __attribute__((ext_vector_type(2)));
typedef float v8f __attribute__((ext_vector_type(8)));

#define NNODES 20000
#define NEDGES 320000
#define NTILES (NNODES / 16)   // 1250, exact

__device__ __constant__ const float LIN1_NORM  = 0.17677669529663687f; // 1/sqrt(32)
__device__ __constant__ const float SC_NORM    = 0.08838834764831843f; // 1/sqrt(128)
__device__ __constant__ const float LIN2_NORM  = 0.125f;               // 1/sqrt(64)
__device__ __constant__ const float EDGE_SCALE = 0.25f;                // 1/sqrt(16)
__device__ __constant__ const float INV_SQRT3  = 0.57735026918962576f;

__device__ __forceinline__ v8f wmma4(v2f a, v2f b, v8f c) {
  // D = A(16x4,f32) * B(4x16,f32) + C(16x16,f32)
  return __builtin_amdgcn_wmma_f32_16x16x4_f32(
      /*neg_a=*/false, a, /*neg_b=*/false, b,
      /*c_mod=*/(short)0, c, /*reuse_a=*/false, /*reuse_b=*/false);
}

// ---------------------------------------------------------------------------
// Kernel 0: zero the scatter accumulator (d_ws is poisoned by the harness).
// ---------------------------------------------------------------------------
__global__ void k_zero(float* __restrict__ p, size_t n) {
  size_t i = (size_t)blockIdx.x * blockDim.x + threadIdx.x;
  const size_t stride = (size_t)gridDim.x * blockDim.x;
  for (; i < n; i += stride) p[i] = 0.0f;
}

// ---------------------------------------------------------------------------
// Kernel 1: per 16-node tile (one wave each):
//   y  = [x0@W10 | x1_m@W11] / sqrt(32)           -> ws (N,128), gather-friendly
//   sc = species-gated GEMMs with sc_w0/sc_w1     -> d_out (N,128)
// x tile (16x128) staged in LDS with padded stride (132 % 64 = 4 -> no bank
// conflicts on the 16-row fragment reads).
// ---------------------------------------------------------------------------
__global__ __launch_bounds__(128) void k_node(
    const float* __restrict__ nf,   const float* __restrict__ attrs,
    const float* __restrict__ W10,  const float* __restrict__ W11,
    const float* __restrict__ scw0, const float* __restrict__ scw1,
    float* __restrict__ y, float* __restrict__ out)
{
  __shared__ float lds[4][16][132];
  const int lane = threadIdx.x & 31;
  const int wave = threadIdx.x >> 5;
  const int tile = blockIdx.x * 4 + wave;
  const bool active = (tile < NTILES);
  const int node0 = tile * 16;

  if (active) {
    // Stage 16 rows x 128 floats, coalesced float4 loads.
    for (int idx = lane; idx < 16 * 32; idx += 32) {
      const int r = idx >> 5;
      const int c = idx & 31;
      const float4 v =
          *reinterpret_cast<const float4*>(nf + (size_t)(node0 + r) * 128 + c * 4);
      lds[wave][r][c * 4 + 0] = v.x;
      lds[wave][r][c * 4 + 1] = v.y;
      lds[wave][r][c * 4 + 2] = v.z;
      lds[wave][r][c * 4 + 3] = v.w;
    }
  }
  __syncthreads();
  if (!active) return;

  const int fm    = lane & 15;          // A row / B col / D col for this lane
  const int fko   = (lane >> 4) << 1;   // K sub-offset: 0 or 2
  const int rbase = (lane >> 4) << 3;   // D row base: 0 or 8

  float am[4];
  #pragma unroll
  for (int s = 0; s < 4; ++s) am[s] = attrs[(size_t)(node0 + fm) * 4 + s];

  // ---- scalar channel (x0 = cols 0..31): y0 and sc0 ----
  #pragma unroll
  for (int t = 0; t < 2; ++t) {
    v8f cy = {};
    v8f cs = {};
    #pragma unroll
    for (int k0 = 0; k0 < 32; k0 += 4) {
      v2f a;
      a.x = lds[wave][fm][k0 + fko];
      a.y = lds[wave][fm][k0 + fko + 1];
      v2f b;
      b.x = W10[(k0 + fko) * 32 + t * 16 + fm];
      b.y = W10[(k0 + fko + 1) * 32 + t * 16 + fm];
      cy = wmma4(a, b, cy);
      #pragma unroll
      for (int s = 0; s < 4; ++s) {
        v2f as = a * am[s];   // one-hot species gate on A rows
        v2f bs;
        bs.x = scw0[(k0 + fko) * 128 + s * 32 + t * 16 + fm];
        bs.y = scw0[(k0 + fko + 1) * 128 + s * 32 + t * 16 + fm];
        cs = wmma4(as, bs, cs);
      }
    }
    #pragma unroll
    for (int r = 0; r < 8; ++r) {
      const int row = node0 + rbase + r;
      const int col = t * 16 + fm;
      y[(size_t)row * 128 + col]   = cy[r] * LIN1_NORM;
      out[(size_t)row * 128 + col] = cs[r] * SC_NORM;
    }
  }

  // ---- vector channels x1[:,:,m] (col 32 + i*3 + m): y1 and sc1 ----
  #pragma unroll
  for (int m = 0; m < 3; ++m) {
    #pragma unroll
    for (int t = 0; t < 2; ++t) {
      v8f cy = {};
      v8f cs = {};
      #pragma unroll
      for (int k0 = 0; k0 < 32; k0 += 4) {
        v2f a;
        a.x = lds[wave][fm][32 + (k0 + fko) * 3 + m];
        a.y = lds[wave][fm][32 + (k0 + fko + 1) * 3 + m];
        v2f b;
        b.x = W11[(k0 + fko) * 32 + t * 16 + fm];
        b.y = W11[(k0 + fko + 1) * 32 + t * 16 + fm];
        cy = wmma4(a, b, cy);
        #pragma unroll
        for (int s = 0; s < 4; ++s) {
          v2f as = a * am[s];
          v2f bs;
          bs.x = scw1[(k0 + fko) * 128 + s * 32 + t * 16 + fm];
          bs.y = scw1[(k0 + fko + 1) * 128 + s * 32 + t * 16 + fm];
          cs = wmma4(as, bs, cs);
        }
      }
      #pragma unroll
      for (int r = 0; r < 8; ++r) {
        const int row = node0 + rbase + r;
        const int col = 32 + (t * 16 + fm) * 3 + m;
        y[(size_t)row * 128 + col]   = cy[r] * LIN1_NORM;
        out[(size_t)row * 128 + col] = cs[r] * SC_NORM;
      }
    }
  }
}

// ---------------------------------------------------------------------------
// Kernel 2: one wave per edge, lane = channel 0..31.
// Edge MLP (silu) in registers, gather y[src] (L2-resident), 8 f32 atomics
// per lane into the (N,256) accumulator: [a0(64) | a1(64x3 i-major)].
// ---------------------------------------------------------------------------
__global__ __launch_bounds__(256) void k_edge(
    const float* __restrict__ emb, const float* __restrict__ eattr,
    const int*  __restrict__ eidx, const float* __restrict__ mw1,
    const float* __restrict__ mw2, const float* __restrict__ y,
    float* __restrict__ acc)
{
  const int lane = threadIdx.x & 31;
  const int e = blockIdx.x * 8 + (threadIdx.x >> 5);
  if (e >= NEDGES) return;

  const int dst = eidx[e];            // edge_index[0]
  const int src = eidx[NEDGES + e];   // edge_index[1]

  float er[8];
  #pragma unroll
  for (int j = 0; j < 8; ++j) er[j] = emb[(size_t)e * 8 + j];

  float h[8];
  #pragma unroll
  for (int k = 0; k < 8; ++k) {
    float s = 0.0f;
    #pragma unroll
    for (int j = 0; j < 8; ++j) s += er[j] * mw1[j * 8 + k];
    h[k] = s / (1.0f + __expf(-s));   // silu
  }
  float w0 = 0.f, w1 = 0.f, w2 = 0.f, w3 = 0.f;
  #pragma unroll
  for (int k = 0; k < 8; ++k) {
    const float hk = h[k];
    w0 += hk * mw2[k * 128 +       lane];
    w1 += hk * mw2[k * 128 +  32 + lane];
    w2 += hk * mw2[k * 128 +  64 + lane];
    w3 += hk * mw2[k * 128 +  96 + lane];
  }

  const float sh0 = eattr[(size_t)e * 4 + 0];
  const float s1x = eattr[(size_t)e * 4 + 1];
  const float s1y = eattr[(size_t)e * 4 + 2];
  const float s1z = eattr[(size_t)e * 4 + 3];

  const float* yr = y + (size_t)src * 128;
  const float e0  = yr[lane];
  const float e1x = yr[32 + lane * 3 + 0];
  const float e1y = yr[32 + lane * 3 + 1];
  const float e1z = yr[32 + lane * 3 + 2];

  float* ar = acc + (size_t)dst * 256;
  const float m0a = w0 * e0 * sh0 * EDGE_SCALE;
  const float dot = e1x * s1x + e1y * s1y + e1z * s1z;
  const float m0b = w3 * dot * (INV_SQRT3 * EDGE_SCALE);
  atomicAdd(ar + lane,      m0a);
  atomicAdd(ar + 32 + lane, m0b);

  const float c1 = w1 * e0 * EDGE_SCALE;        // m1a channel
  atomicAdd(ar + 64 + lane * 3 + 0, c1 * s1x);
  atomicAdd(ar + 64 + lane * 3 + 1, c1 * s1y);
  atomicAdd(ar + 64 + lane * 3 + 2, c1 * s1z);

  const float c2 = w2 * sh0 * EDGE_SCALE;       // m1b channel
  atomicAdd(ar + 64 + (32 + lane) * 3 + 0, c2 * e1x);
  atomicAdd(ar + 64 + (32 + lane) * 3 + 1, c2 * e1y);
  atomicAdd(ar + 64 + (32 + lane) * 3 + 2, c2 * e1z);
}

// ---------------------------------------------------------------------------
// Kernel 3: per 16-node tile: o0 = a0 @ W20, o1_m = a1[:,:,m] @ W21 (K=64),
// scaled by 1/sqrt(64) and accumulated onto the sc values already in d_out.
// ---------------------------------------------------------------------------
__global__ __launch_bounds__(256) void k_out(
    const float* __restrict__ acc, const float* __restrict__ W20,
    const float* __restrict__ W21, float* __restrict__ out)
{
  const int lane = threadIdx.x & 31;
  const int tile = blockIdx.x * 8 + (threadIdx.x >> 5);
  if (tile >= NTILES) return;
  const int node0 = tile * 16;

  const int fm    = lane & 15;
  const int fko   = (lane >> 4) << 1;
  const int rbase = (lane >> 4) << 3;
  const float* arow = acc + (size_t)(node0 + fm) * 256;

  // ---- o0 ----
  #pragma unroll
  for (int t = 0; t < 2; ++t) {
    v8f c = {};
    #pragma unroll
    for (int k0 = 0; k0 < 64; k0 += 4) {
      v2f a, b;
      a.x = arow[k0 + fko];
      a.y = arow[k0 + fko + 1];
      b.x = W20[(k0 + fko) * 32 + t * 16 + fm];
      b.y = W20[(k0 + fko + 1) * 32 + t * 16 + fm];
      c = wmma4(a, b, c);
    }
    #pragma unroll
    for (int r = 0; r < 8; ++r) {
      const size_t o = (size_t)(node0 + rbase + r) * 128 + t * 16 + fm;
      out[o] += c[r] * LIN2_NORM;
    }
  }

  // ---- o1 per vector component m ----
  #pragma unroll
  for (int m = 0; m < 3; ++m) {
    #pragma unroll
    for (int t = 0; t < 2; ++t) {
      v8f c = {};
      #pragma unroll
      for (int k0 = 0; k0 < 64; k0 += 4) {
        v2f a, b;
        a.x = arow[64 + (k0 + fko) * 3 + m];
        a.y = arow[64 + (k0 + fko + 1) * 3 + m];
        b.x = W21[(k0 + fko) * 32 + t * 16 + fm];
        b.y = W21[(k0 + fko + 1) * 32 + t * 16 + fm];
        c = wmma4(a, b, c);
      }
      #pragma unroll
      for (int r = 0; r < 8; ++r) {
        const size_t o =
            (size_t)(node0 + rbase + r) * 128 + 32 + (t * 16 + fm) * 3 + m;
        out[o] += c[r] * LIN2_NORM;
      }
    }
  }
}

// ---------------------------------------------------------------------------
extern "C" void kernel_launch(void* const* d_in, const int* in_sizes, int n_in,
                              void* d_out, int out_size, void* d_ws, size_t ws_size,
                              hipStream_t stream)
{
  const float* nf   = (const float*)d_in[0];   // node_features (N,128)
  const float* attrs= (const float*)d_in[1];   // node_attrs (N,4)
  const float* emb  = (const float*)d_in[2];   // edge_embedding (E,8)
  const float* eat  = (const float*)d_in[3];   // edge_attrs (E,4)
  const int*   eidx = (const int*)d_in[4];     // edge_index (2,E)
  const float* W10  = (const float*)d_in[5];
  const float* W11  = (const float*)d_in[6];
  const float* mw1  = (const float*)d_in[7];
  const float* mw2  = (const float*)d_in[8];
  const float* W20  = (const float*)d_in[9];
  const float* W21  = (const float*)d_in[10];
  const float* scw0 = (const float*)d_in[11];
  const float* scw1 = (const float*)d_in[12];

  float* out = (float*)d_out;
  float* y   = (float*)d_ws;                       // (N,128)
  float* acc = y + (size_t)NNODES * 128;           // (N,256)

  k_zero<<<2048, 256, 0, stream>>>(acc, (size_t)NNODES * 256);
  k_node<<<(NTILES + 3) / 4, 128, 0, stream>>>(nf, attrs, W10, W11, scw0, scw1, y, out);
  k_edge<<<NEDGES / 8, 256, 0, stream>>>(emb, eat, eidx, mw1, mw2, y, acc);
  k_out<<<(NTILES + 7) / 8, 256, 0, stream>>>(acc, W20, W21, out);
}